// NoisyTopkRouter_83803401880037
// MI455X (gfx1250) — compile-verified
//
#include <hip/hip_runtime.h>

typedef float v2f __attribute__((ext_vector_type(2)));
typedef float v8f __attribute__((ext_vector_type(8)));

#define D_MODEL      1024
#define NUM_EXPERTS  64
#define TOKENS_PER_WG 128
#define THREADS      128   // 4 waves (wave32)

// Each wave: 32 tokens (2 M-tiles of 16) x 64 experts (4 N-tiles of 16),
// for both W_route and W_noise, via V_WMMA_F32_16X16X4_F32.
__global__ __launch_bounds__(THREADS)
void noisy_topk_router_kernel(const float* __restrict__ mh,
                              const float* __restrict__ noise,
                              const float* __restrict__ Wr,
                              const float* __restrict__ br,
                              const float* __restrict__ Wn,
                              const float* __restrict__ bn,
                              float* __restrict__ out_router,
                              int*   __restrict__ out_idx)
{
    // +1 column pad so the epilogue's column-wise reads hit distinct banks
    __shared__ float s_logit[TOKENS_PER_WG][NUM_EXPERTS + 1];
    __shared__ float s_nlog [TOKENS_PER_WG][NUM_EXPERTS + 1];

    const int tid    = threadIdx.x;
    const int wave   = tid >> 5;
    const int lane   = tid & 31;
    const int lane_m = lane & 15;          // M (A) / N (B) index within tile
    const int k_half = (lane >> 4) << 1;   // K offset: 0 for lanes 0-15, 2 for 16-31

    const int wg_tok = blockIdx.x * TOKENS_PER_WG;
    const int row0   = wg_tok + wave * 32 + lane_m;

    // ---- accumulators, initialized with bias (C[m][n] = b[n]) ----
    v8f accR[2][4];
    v8f accN[2][4];
#pragma unroll
    for (int nc = 0; nc < 4; ++nc) {
        const float brv = br[nc * 16 + lane_m];
        const float bnv = bn[nc * 16 + lane_m];
#pragma unroll
        for (int t = 0; t < 2; ++t) {
#pragma unroll
            for (int v = 0; v < 8; ++v) {
                accR[t][nc][v] = brv;
                accN[t][nc][v] = bnv;
            }
        }
    }

    // ---- fragment base pointers ----
    const float* a0p = mh + (size_t)row0 * D_MODEL + k_half;
    const float* a1p = a0p + (size_t)16 * D_MODEL;
    const float* brp[4];
    const float* bnp[4];
#pragma unroll
    for (int nc = 0; nc < 4; ++nc) {
        const int e = nc * 16 + lane_m;
        brp[nc] = Wr + (size_t)e * D_MODEL + k_half;
        bnp[nc] = Wn + (size_t)e * D_MODEL + k_half;
    }

    // ---- K loop: 256 steps of K=4 ----
#pragma unroll 4
    for (int k = 0; k < D_MODEL; k += 4) {
        const v2f a0 = *(const v2f*)(a0p + k);
        const v2f a1 = *(const v2f*)(a1p + k);
#pragma unroll
        for (int nc = 0; nc < 4; ++nc) {
            const v2f bR = *(const v2f*)(brp[nc] + k);
            const v2f bN = *(const v2f*)(bnp[nc] + k);
            accR[0][nc] = __builtin_amdgcn_wmma_f32_16x16x4_f32(
                false, a0, false, bR, (short)0, accR[0][nc], false, false);
            accR[1][nc] = __builtin_amdgcn_wmma_f32_16x16x4_f32(
                false, a1, false, bR, (short)0, accR[1][nc], false, false);
            accN[0][nc] = __builtin_amdgcn_wmma_f32_16x16x4_f32(
                false, a0, false, bN, (short)0, accN[0][nc], false, false);
            accN[1][nc] = __builtin_amdgcn_wmma_f32_16x16x4_f32(
                false, a1, false, bN, (short)0, accN[1][nc], false, false);
        }
    }

    // ---- spill C tiles to LDS (D layout: lanes 0-15 -> M=v, 16-31 -> M=v+8) ----
    const int m_base = (lane < 16) ? 0 : 8;
#pragma unroll
    for (int t = 0; t < 2; ++t) {
#pragma unroll
        for (int nc = 0; nc < 4; ++nc) {
            const int col   = nc * 16 + lane_m;
            const int rbase = wave * 32 + t * 16 + m_base;
#pragma unroll
            for (int v = 0; v < 8; ++v) {
                s_logit[rbase + v][col] = accR[t][nc][v];
                s_nlog [rbase + v][col] = accN[t][nc][v];
            }
        }
    }
    __syncthreads();

    // ---- epilogue: one token per thread ----
    const int   tok  = wg_tok + tid;
    const float* npr = noise + (size_t)tok * NUM_EXPERTS;

    const float NEG_INF = -__builtin_inff();
    float v1 = NEG_INF, v2 = NEG_INF;
    int   i1 = 0,        i2 = 0;
#pragma unroll 4
    for (int e = 0; e < NUM_EXPERTS; ++e) {
        const float lg = s_logit[tid][e];
        const float nl = s_nlog [tid][e];
        // numerically stable softplus: max(x,0) + log1p(exp(-|x|))
        const float sp = fmaxf(nl, 0.0f) + log1pf(expf(-fabsf(nl)));
        const float x  = lg + npr[e] * sp;
        if (x > v1)      { v2 = v1; i2 = i1; v1 = x; i1 = e; }
        else if (x > v2) { v2 = x;  i2 = e; }
    }

    // softmax over the top-2 values (v1 >= v2)
    const float ee = expf(v2 - v1);
    const float inv = 1.0f / (1.0f + ee);
    const float p1 = inv;
    const float p2 = ee * inv;

    float4* orow = (float4*)(out_router + (size_t)tok * NUM_EXPERTS);
#pragma unroll
    for (int q = 0; q < NUM_EXPERTS / 4; ++q) {
        const int e = q * 4;
        float4 w;
        w.x = (e + 0 == i1) ? p1 : ((e + 0 == i2) ? p2 : 0.0f);
        w.y = (e + 1 == i1) ? p1 : ((e + 1 == i2) ? p2 : 0.0f);
        w.z = (e + 2 == i1) ? p1 : ((e + 2 == i2) ? p2 : 0.0f);
        w.w = (e + 3 == i1) ? p1 : ((e + 3 == i2) ? p2 : 0.0f);
        orow[q] = w;
    }
    out_idx[2 * tok]     = i1;
    out_idx[2 * tok + 1] = i2;
}

extern "C" void kernel_launch(void* const* d_in, const int* in_sizes, int n_in,
                              void* d_out, int out_size, void* d_ws, size_t ws_size,
                              hipStream_t stream) {
    const float* mh    = (const float*)d_in[0];  // [8,4096,1024]
    const float* noise = (const float*)d_in[1];  // [8,4096,64]
    const float* Wr    = (const float*)d_in[2];  // [64,1024]
    const float* br    = (const float*)d_in[3];  // [64]
    const float* Wn    = (const float*)d_in[4];  // [64,1024]
    const float* bn    = (const float*)d_in[5];  // [64]

    const int n_tokens = in_sizes[0] / D_MODEL;  // 32768

    float* out_router = (float*)d_out;                                   // [tokens,64] f32
    int*   out_idx    = (int*)d_out + (size_t)n_tokens * NUM_EXPERTS;    // [tokens,2] i32

    dim3 grid(n_tokens / TOKENS_PER_WG);
    dim3 block(THREADS);
    hipLaunchKernelGGL(noisy_topk_router_kernel, grid, block, 0, stream,
                       mh, noise, Wr, br, Wn, bn, out_router, out_idx);
}